// Portilla_Simoncelli_66718021976087
// MI455X (gfx1250) — compile-verified
//
#include <hip/hip_runtime.h>
#include <math.h>

// ============================ constants ============================
#define PS_NS 4
#define PS_K  4

typedef __attribute__((ext_vector_type(2))) float v2f;
typedef __attribute__((ext_vector_type(8))) float v8f;

// output layout offsets (float32 elements)
#define O_PIX  0      // 6
#define O_MM   6      // 18
#define O_ACM  24     // 9*9*4*4 = 1296
#define O_SK   1320   // 5
#define O_KU   1325   // 5
#define O_AC   1330   // 9*9*5 = 405
#define O_COCM 1735   // 4*4*5 = 80
#define O_CSCM 1815   // 4*4*4 = 64
#define O_COCR 1879   // 8*8*5 = 320
#define O_CSCR 2199   // 8*8*4 = 256
#define O_VHP  2455   // 1
#define O_TOT  2456

// STATS slots (floats / aliased unsigned)
#define ST_SUM  0
#define ST_SUM2 1
#define ST_SUM3 2
#define ST_SUM4 3
#define ST_MINK 4
#define ST_MAXK 5
#define ST_VAR0 8
#define ST_SCAL 9
#define ST_VARI 10

#define PI_F 3.14159265358979f
#define TWO_PI_F 6.28318530717959f

// ============================ device helpers ============================
__device__ __forceinline__ unsigned mapf(float x) {
  unsigned u = __float_as_uint(x);
  return (u & 0x80000000u) ? ~u : (u | 0x80000000u);
}
__device__ __forceinline__ float unmapf(unsigned k) {
  return __uint_as_float((k & 0x80000000u) ? (k & 0x7FFFFFFFu) : ~k);
}

// frequency coordinate for UNSHIFTED spectrum index k (N power of two).
// mask value at unshifted k == centered mask at index (k+N/2)%N.
__device__ __forceinline__ float freq_coord(int k, int N) {
  int i = (k + (N >> 1)) & (N - 1);
  return (float)(i - (N >> 1)) * (2.0f / (float)N);
}
__device__ __forceinline__ void polar_at(int ky, int kx, int N, float* lr, float* ang) {
  float fy = freq_coord(ky, N), fx = freq_coord(kx, N);
  float r = sqrtf(fx * fx + fy * fy);
  if (r == 0.0f) r = 1.0f / (float)N;
  *lr = log2f(r);
  *ang = atan2f(fy, fx);
}
__device__ __forceinline__ float rc_hi_f(float lr, float lo, float hi) {
  float t = (lr - lo) / (hi - lo);
  t = fminf(fmaxf(t, 0.0f), 1.0f);
  return sqrtf(0.5f * (1.0f - cosf(PI_F * t)));
}
__device__ __forceinline__ float rc_lo_f(float lr, float lo, float hi) {
  float t = (lr - lo) / (hi - lo);
  t = fminf(fmaxf(t, 0.0f), 1.0f);
  return sqrtf(0.5f * (1.0f + cosf(PI_F * t)));
}
// K=4, order=3, const = 2^6*(3!)^2/(4*6!) = 0.8
__device__ __forceinline__ float angle_mask_f(float ang, int b, int is_complex) {
  float d = fmodf(ang - PI_F * (float)b / 4.0f + PI_F, TWO_PI_F);
  if (d < 0.0f) d += TWO_PI_F;
  d -= PI_F;
  float c = cosf(d);
  float m = 0.894427191f * c * c * c;   // sqrt(0.8)*cos^3
  if (is_complex) m *= 2.0f * ((fabsf(d) < (0.5f * PI_F)) ? 1.0f : 0.0f);
  return m;
}

// ============================ FFT (LDS radix-2) ============================
#define FFT_T 256
__global__ void k_fft1d(const float2* in, float2* out, int N, int logN,
                        int colMode, float sign, int norm) {
  __shared__ float re[1024];
  __shared__ float im[1024];
  const int line = blockIdx.x;
  const int tid = threadIdx.x;
  for (int t = tid; t < N; t += FFT_T) {
    int g = colMode ? (t * N + line) : (line * N + t);
    unsigned r = __brev((unsigned)t) >> (32 - logN);
    float2 v = in[g];
    re[r] = v.x;
    im[r] = v.y;
  }
  __syncthreads();
  for (int st = 1; st <= logN; ++st) {
    int m = 1 << st, half = m >> 1;
    for (int j = tid; j < (N >> 1); j += FFT_T) {
      int pos = j & (half - 1);
      int grp = j >> (st - 1);
      int i0 = grp * m + pos, i1 = i0 + half;
      float angv = sign * TWO_PI_F * (float)pos / (float)m;
      float wi, wr;
      __sincosf(angv, &wi, &wr);
      float xr = re[i1], xi = im[i1];
      float tr = xr * wr - xi * wi;
      float ti = xr * wi + xi * wr;
      float ur = re[i0], ui = im[i0];
      re[i0] = ur + tr; im[i0] = ui + ti;
      re[i1] = ur - tr; im[i1] = ui - ti;
    }
    __syncthreads();
  }
  float sc = norm ? (1.0f / (float)N) : 1.0f;
  for (int t = tid; t < N; t += FFT_T) {
    int g = colMode ? (t * N + line) : (line * N + t);
    out[g] = make_float2(re[t] * sc, im[t] * sc);
  }
}

// ============================ elementwise kernels ============================
__global__ void k_zero(float* p, int n) {
  int i = blockIdx.x * blockDim.x + threadIdx.x;
  if (i < n) p[i] = 0.0f;
}
__global__ void k_pack(const float* in, float2* out, int n) {
  int i = blockIdx.x * blockDim.x + threadIdx.x;
  if (i < n) out[i] = make_float2(in[i], 0.0f);
}
__global__ void k_real(const float2* in, float* out, int n) {
  int i = blockIdx.x * blockDim.x + threadIdx.x;
  if (i < n) out[i] = in[i].x;
}
__global__ void k_real_add(const float2* in, float* out, int n) {
  int i = blockIdx.x * blockDim.x + threadIdx.x;
  if (i < n) out[i] += in[i].x;
}
__global__ void k_mag(const float2* in, float* out, int n) {
  int i = blockIdx.x * blockDim.x + threadIdx.x;
  if (i < n) { float x = in[i].x, y = in[i].y; out[i] = sqrtf(x * x + y * y); }
}
__global__ void k_power(const float2* in, float2* out, int n) {
  int i = blockIdx.x * blockDim.x + threadIdx.x;
  if (i < n) { float x = in[i].x, y = in[i].y; out[i] = make_float2(x * x + y * y, 0.0f); }
}
__global__ void k_sub_scalar(float* p, const float* stats, int n) {
  int i = blockIdx.x * blockDim.x + threadIdx.x;
  if (i < n) p[i] -= stats[ST_SCAL];
}

// ============================ mask kernels ============================
__global__ void k_mask_hi0(const float2* in, float2* out, int N) {
  int i = blockIdx.x * blockDim.x + threadIdx.x;
  if (i >= N * N) return;
  float lr, ang; polar_at(i / N, i % N, N, &lr, &ang);
  float m = rc_hi_f(lr, -1.0f, 0.0f);
  out[i] = make_float2(in[i].x * m, in[i].y * m);
}
__global__ void k_mask_lo0(const float2* in, float2* out, int N) {
  int i = blockIdx.x * blockDim.x + threadIdx.x;
  if (i >= N * N) return;
  float lr, ang; polar_at(i / N, i % N, N, &lr, &ang);
  float m = rc_lo_f(lr, -1.0f, 0.0f);
  out[i] = make_float2(in[i].x * m, in[i].y * m);
}
// band = cph * lodft * rc_hi(-2,-1) * am_complex(b);  cph=(-i)^3 = +i
__global__ void k_band_mask(const float2* in, float2* out, int N, int b) {
  int i = blockIdx.x * blockDim.x + threadIdx.x;
  if (i >= N * N) return;
  float lr, ang; polar_at(i / N, i % N, N, &lr, &ang);
  float m = rc_hi_f(lr, -2.0f, -1.0f) * angle_mask_f(ang, b, 1);
  float xr = in[i].x * m, xi = in[i].y * m;
  out[i] = make_float2(-xi, xr);          // multiply by +i
}
// recondft += (i^3 = -i) * F * rc_hi(-2,-1) * am_real(b)
__global__ void k_recon_accum(const float2* in, float2* acc, int N, int b) {
  int i = blockIdx.x * blockDim.x + threadIdx.x;
  if (i >= N * N) return;
  float lr, ang; polar_at(i / N, i % N, N, &lr, &ang);
  float m = rc_hi_f(lr, -2.0f, -1.0f) * angle_mask_f(ang, b, 0);
  float xr = in[i].x * m, xi = in[i].y * m;
  acc[i].x += xi;                          // (x+iy)(-i) = y - ix
  acc[i].y += -xr;
}
// crop centered half of spectrum (unshifted layout): Nout = Nin/2
__global__ void k_crop(const float2* in, float2* out, int Nout) {
  int i = blockIdx.x * blockDim.x + threadIdx.x;
  if (i >= Nout * Nout) return;
  int Nin = Nout << 1;
  int ky = i / Nout, kx = i % Nout;
  int fy = (ky < (Nout >> 1)) ? ky : ky - Nout;
  int fx = (kx < (Nout >> 1)) ? kx : kx - Nout;
  int iy = (fy >= 0) ? fy : fy + Nin;
  int ix = (fx >= 0) ? fx : fx + Nin;
  out[i] = in[iy * Nin + ix];
}
// zero-pad spectrum Nin -> 2*Nin (net x4/ /4 factors cancel in usage)
__global__ void k_expand(const float2* in, float2* out, int Nin) {
  int Nout = Nin << 1;
  int i = blockIdx.x * blockDim.x + threadIdx.x;
  if (i >= Nout * Nout) return;
  int ky = i / Nout, kx = i % Nout;
  int fy = (ky < (Nout >> 1)) ? ky : ky - Nout;
  int fx = (kx < (Nout >> 1)) ? kx : kx - Nout;
  float2 v = make_float2(0.0f, 0.0f);
  if (fy >= -(Nin >> 1) && fy <= (Nin >> 1) - 1 &&
      fx >= -(Nin >> 1) && fx <= (Nin >> 1) - 1) {
    int iy = (fy >= 0) ? fy : fy + Nin;
    int ix = (fx >= 0) ? fx : fx + Nin;
    v = in[iy * Nin + ix];
  }
  out[i] = v;
}

// ============================ reductions / finalize ============================
__global__ void k_stats_init(float* stats) {
  stats[ST_SUM] = 0.0f; stats[ST_SUM2] = 0.0f;
  stats[ST_SUM3] = 0.0f; stats[ST_SUM4] = 0.0f;
  ((unsigned*)stats)[ST_MINK] = 0xFFFFFFFFu;
  ((unsigned*)stats)[ST_MAXK] = 0u;
}
__global__ void k_reduce(const float* in, int n, int useAbs, float* stats) {
  __shared__ float sh0[256], sh1[256], sh2[256], sh3[256];
  __shared__ unsigned shm[256], shx[256];
  int tid = threadIdx.x;
  float a0 = 0, a1 = 0, a2 = 0, a3 = 0;
  unsigned mn = 0xFFFFFFFFu, mx = 0u;
  for (long i = (long)blockIdx.x * blockDim.x + tid; i < n;
       i += (long)gridDim.x * blockDim.x) {
    float x = in[i];
    if (useAbs) x = fabsf(x);
    float x2 = x * x;
    a0 += x; a1 += x2; a2 += x2 * x; a3 += x2 * x2;
    unsigned k = mapf(x);
    mn = (k < mn) ? k : mn;
    mx = (k > mx) ? k : mx;
  }
  sh0[tid] = a0; sh1[tid] = a1; sh2[tid] = a2; sh3[tid] = a3;
  shm[tid] = mn; shx[tid] = mx;
  __syncthreads();
  for (int s = 128; s > 0; s >>= 1) {
    if (tid < s) {
      sh0[tid] += sh0[tid + s]; sh1[tid] += sh1[tid + s];
      sh2[tid] += sh2[tid + s]; sh3[tid] += sh3[tid + s];
      shm[tid] = (shm[tid + s] < shm[tid]) ? shm[tid + s] : shm[tid];
      shx[tid] = (shx[tid + s] > shx[tid]) ? shx[tid + s] : shx[tid];
    }
    __syncthreads();
  }
  if (tid == 0) {
    atomicAdd(&stats[ST_SUM], sh0[0]);
    atomicAdd(&stats[ST_SUM2], sh1[0]);
    atomicAdd(&stats[ST_SUM3], sh2[0]);
    atomicAdd(&stats[ST_SUM4], sh3[0]);
    atomicMin((unsigned*)stats + ST_MINK, shm[0]);
    atomicMax((unsigned*)stats + ST_MAXK, shx[0]);
  }
}
__global__ void k_fin_pix(float* stats, float* out, float n) {
  float s = stats[ST_SUM], s2 = stats[ST_SUM2], s3 = stats[ST_SUM3], s4 = stats[ST_SUM4];
  float mu = s / n;
  float m2r = s2 / n, m3r = s3 / n, m4r = s4 / n;
  float var0 = (s2 - n * mu * mu) / (n - 1.0f);
  float m3 = m3r - 3.0f * mu * m2r + 2.0f * mu * mu * mu;
  float m4 = m4r - 4.0f * mu * m3r + 6.0f * mu * mu * m2r - 3.0f * mu * mu * mu * mu;
  out[O_PIX + 0] = mu;
  out[O_PIX + 1] = var0;
  out[O_PIX + 2] = m3 / powf(var0, 1.5f);
  out[O_PIX + 3] = m4 / (var0 * var0);
  out[O_PIX + 4] = unmapf(((unsigned*)stats)[ST_MINK]);
  out[O_PIX + 5] = unmapf(((unsigned*)stats)[ST_MAXK]);
  stats[ST_VAR0] = var0;
}
__global__ void k_fin_mean(float* stats, float* out, float n, int outIdx) {
  float mean = stats[ST_SUM] / n;
  stats[ST_SCAL] = mean;
  if (outIdx >= 0) out[outIdx] = mean;
}
__global__ void k_fin_skewkurt(const float* stats, float* out, float n, int s) {
  float vari = stats[ST_VARI], var0 = stats[ST_VAR0];
  int cond = (vari / var0) > 1e-6f;
  out[O_SK + s] = cond ? (stats[ST_SUM3] / n) / powf(vari, 1.5f) : 0.0f;
  out[O_KU + s] = cond ? (stats[ST_SUM4] / n) / (vari * vari) : 3.0f;
}
__global__ void k_fin_varhp(const float* stats, float* out, float n) {
  out[O_VHP] = stats[ST_SUM2] / n;
}

// extract fftshifted 9x9 autocorr patch; ac already ifft2-normalized; extra /(N*N)
__global__ void k_extract_ac(const float2* ac, int N, float* out,
                             int baseOut, int strideOut, float* stats) {
  int idx = threadIdx.x;
  if (idx >= 81) return;
  int py = idx / 9, px = idx % 9;
  int dy = py - 4, dx = px - 4;
  int ry = ((dy % N) + N) % N;
  int rx = ((dx % N) + N) % N;
  float v = ac[ry * N + rx].x / ((float)N * (float)N);
  out[baseOut + (py * 9 + px) * strideOut] = v;
  if (dy == 0 && dx == 0) stats[ST_VARI] = v;
}

// ============================ parents helpers ============================
__global__ void k_parent_cols(const float2* c, const float* stats,
                              float* parm, float* rpr, float* rpi, int n) {
  int i = blockIdx.x * blockDim.x + threadIdx.x;
  if (i >= n) return;
  float rt = c[i].x, it = c[i].y;
  float mg = sqrtf(rt * rt + it * it);
  float a2 = 2.0f * atan2f(rt, it);
  parm[i] = mg - stats[ST_SCAL];
  rpr[i] = mg * cosf(a2);
  rpi[i] = mg * sinf(a2);
}
// 5 shifted copies (lowpass parents); transposed-flatten permutation is common
// to all columns so we emit in original (y,x) order with swapped-axis rolls.
__global__ void k_roll5(const float* t, float* R, int N) {
  int n = N * N;
  int i = blockIdx.x * blockDim.x + threadIdx.x;
  if (i >= n) return;
  int y = i / N, x = i % N;
  R[0 * n + i] = t[i];
  R[1 * n + i] = t[y * N + ((x - 1 + N) % N)];
  R[2 * n + i] = t[y * N + ((x + 1) % N)];
  R[3 * n + i] = t[((y - 1 + N) % N) * N + x];
  R[4 * n + i] = t[((y + 1) % N) * N + x];
}

// ============================ WMMA correlation: C(16x16) += A^T B ============================
// A: n x pA (column-major, col*n+i), B: n x pB. Chunked K=4 via
// V_WMMA_F32_16X16X4_F32, unrolled x4 (16 K-values / iteration).
// Branch-free: out-of-range columns are clamped to column 0 (valid memory)
// and zeroed with a 0/1 multiplier, so loads are unconditional b64 and EXEC
// stays all-ones for the WMMA pipe.
__global__ __launch_bounds__(32) void k_wmma_corr(const float* __restrict__ A,
                                                  const float* __restrict__ B,
                                                  int n, int pA, int pB,
                                                  float* Cacc) {
  const int lane = threadIdx.x & 31;
  const int idx16 = lane & 15;
  const int kb = (lane >> 4) << 1;                  // K base 0 or 2
  const float ma = (idx16 < pA) ? 1.0f : 0.0f;
  const float mb = (idx16 < pB) ? 1.0f : 0.0f;
  const int ca = (idx16 < pA) ? idx16 : 0;
  const int cb = (idx16 < pB) ? idx16 : 0;
  const float* pa = A + (size_t)ca * (size_t)n + kb;
  const float* pb = B + (size_t)cb * (size_t)n + kb;
  v8f c = {0.0f, 0.0f, 0.0f, 0.0f, 0.0f, 0.0f, 0.0f, 0.0f};
  const int step = (int)gridDim.x * 16;             // n is a multiple of 16
  for (int i0 = blockIdx.x * 16; i0 + 15 < n; i0 += step) {
    __builtin_prefetch(pa + i0 + step, 0, 1);
    __builtin_prefetch(pb + i0 + step, 0, 1);
#pragma unroll
    for (int u = 0; u < 4; ++u) {
      const float2 av = *(const float2*)(pa + i0 + 4 * u);
      const float2 bv = *(const float2*)(pb + i0 + 4 * u);
      v2f a; a.x = av.x * ma; a.y = av.y * ma;
      v2f b; b.x = bv.x * mb; b.y = bv.y * mb;
      c = __builtin_amdgcn_wmma_f32_16x16x4_f32(false, a, false, b, (short)0, c,
                                                false, false);
    }
  }
  const int mhi = (lane >> 4) << 3;                 // M offset 0 or 8
#pragma unroll
  for (int v = 0; v < 8; ++v) atomicAdd(&Cacc[(v + mhi) * 16 + idx16], c[v]);
}
__global__ void k_corr_write(const float* Cacc, float* out, int rows, int cols,
                             int Q, int S, int base, float invDiv) {
  int idx = blockIdx.x * blockDim.x + threadIdx.x;
  if (idx >= rows * cols) return;
  int i = idx / cols, j = idx % cols;
  out[base + (i * Q + j) * S] = Cacc[i * 16 + j] * invDiv;
}

// ============================ host orchestration ============================
extern "C" void kernel_launch(void* const* d_in, const int* in_sizes, int n_in,
                              void* d_out_v, int out_size, void* d_ws, size_t ws_size,
                              hipStream_t stream) {
  (void)in_sizes; (void)n_in; (void)out_size; (void)ws_size;
  const float* img = (const float*)d_in[0];
  float* out = (float*)d_out_v;

  const int N0 = 1024, K = PS_K, NS = PS_NS;
  int Nsz[5], S[5];
  for (int s = 0; s <= 4; ++s) { Nsz[s] = N0 >> s; S[s] = Nsz[s] * Nsz[s]; }

  size_t off = 0;
  auto alloc = [&](size_t nfl) {
    void* p = (char*)d_ws + off;
    off += ((nfl + 3) & ~(size_t)3) * sizeof(float);
    return p;
  };
  float2* CTA  = (float2*)alloc(2 * (size_t)S[0]);
  float2* CTB  = (float2*)alloc(2 * (size_t)S[0]);
  float2* LOD  = (float2*)alloc(2 * (size_t)S[0]);
  float2* RDFT = (float2*)alloc(2 * (size_t)S[0]);
  float2* BAND[4][4];
  for (int s = 0; s < 4; ++s)
    for (int b = 0; b < K; ++b) BAND[s][b] = (float2*)alloc(2 * (size_t)S[s]);
  float* MAGd[4];   for (int s = 0; s < 4; ++s) MAGd[s]   = (float*)alloc((size_t)K * S[s]);
  float* REALCd[4]; for (int s = 0; s < 4; ++s) REALCd[s] = (float*)alloc((size_t)K * S[s]);
  float* RHP   = (float*)alloc(S[0]);
  float* RLP   = (float*)alloc(S[4]);
  float* RECON = (float*)alloc(S[0]);
  float* MGT   = (float*)alloc(S[0]);
  float* PARM  = (float*)alloc(4 * (size_t)S[0]);
  float* RPAR  = (float*)alloc(8 * (size_t)S[0]);
  float* STATS = (float*)alloc(64);
  float* CACC  = (float*)alloc(256);

  auto EL = [](int n) { return dim3((unsigned)((n + 255) / 256)); };
  auto fft2 = [&](const float2* in, float2* o, int N, float sign, int norm) {
    int logN = 31 - __builtin_clz((unsigned)N);
    k_fft1d<<<N, FFT_T, 0, stream>>>(in, o, N, logN, 0, sign, norm);
    k_fft1d<<<N, FFT_T, 0, stream>>>(o, o, N, logN, 1, sign, norm);
  };
  auto reduceArr = [&](const float* p, int n, int useAbs) {
    k_stats_init<<<1, 1, 0, stream>>>(STATS);
    k_reduce<<<256, 256, 0, stream>>>(p, n, useAbs, STATS);
  };
  auto autocorr = [&](const float* ch, int N, int baseOut, int strideOut) {
    k_pack<<<EL(N * N), 256, 0, stream>>>(ch, CTA, N * N);
    fft2(CTA, CTA, N, -1.0f, 0);
    k_power<<<EL(N * N), 256, 0, stream>>>(CTA, CTB, N * N);
    fft2(CTB, CTB, N, +1.0f, 1);
    k_extract_ac<<<1, 128, 0, stream>>>(CTB, N, out, baseOut, strideOut, STATS);
  };

  // ---- init output (partially-written tensors must be zero) ----
  k_zero<<<EL(O_TOT), 256, 0, stream>>>(out, O_TOT);

  // ---- pixel stats ----
  reduceArr(img, S[0], 0);
  k_fin_pix<<<1, 1, 0, stream>>>(STATS, out, (float)S[0]);

  // ---- steerable pyramid ----
  k_pack<<<EL(S[0]), 256, 0, stream>>>(img, CTA, S[0]);
  fft2(CTA, CTA, N0, -1.0f, 0);
  k_mask_hi0<<<EL(S[0]), 256, 0, stream>>>(CTA, CTB, N0);
  fft2(CTB, CTB, N0, +1.0f, 1);
  k_real<<<EL(S[0]), 256, 0, stream>>>(CTB, RHP, S[0]);
  k_mask_lo0<<<EL(S[0]), 256, 0, stream>>>(CTA, LOD, N0);

  for (int s = 0; s < NS; ++s) {
    int N = Nsz[s], n = S[s];
    for (int b = 0; b < K; ++b) {
      k_band_mask<<<EL(n), 256, 0, stream>>>(LOD, CTB, N, b);
      fft2(CTB, BAND[s][b], N, +1.0f, 1);
    }
    int Nh = N >> 1;
    k_crop<<<EL(Nh * Nh), 256, 0, stream>>>(LOD, CTB, Nh);
    k_mask_lo0<<<EL(Nh * Nh), 256, 0, stream>>>(CTB, LOD, Nh);
  }
  fft2(LOD, CTB, 64, +1.0f, 1);
  k_real<<<EL(S[4]), 256, 0, stream>>>(CTB, RLP, S[4]);
  reduceArr(RLP, S[4], 0);
  k_fin_mean<<<1, 1, 0, stream>>>(STATS, out, (float)S[4], -1);
  k_sub_scalar<<<EL(S[4]), 256, 0, stream>>>(RLP, STATS, S[4]);

  // ---- magnitude means / centered magnitudes / real parts ----
  reduceArr(RHP, S[0], 1);
  k_fin_mean<<<1, 1, 0, stream>>>(STATS, out, (float)S[0], O_MM + 0);
  for (int s = 0; s < NS; ++s) {
    int n = S[s];
    for (int b = 0; b < K; ++b) {
      float* mg = MAGd[s] + (size_t)b * n;
      k_mag<<<EL(n), 256, 0, stream>>>(BAND[s][b], mg, n);
      reduceArr(mg, n, 0);
      k_fin_mean<<<1, 1, 0, stream>>>(STATS, out, (float)n, O_MM + 1 + s * K + b);
      k_sub_scalar<<<EL(n), 256, 0, stream>>>(mg, STATS, n);
      k_real<<<EL(n), 256, 0, stream>>>(BAND[s][b], REALCd[s] + (size_t)b * n, n);
    }
  }
  reduceArr(RLP, S[4], 1);
  k_fin_mean<<<1, 1, 0, stream>>>(STATS, out, (float)S[4], O_MM + 17);

  // ---- recon chain, autocorrs, skew/kurt ----
  k_pack<<<EL(S[4]), 256, 0, stream>>>(RLP, CTA, S[4]);
  fft2(CTA, CTA, 64, -1.0f, 0);
  k_mask_lo0<<<EL(S[4]), 256, 0, stream>>>(CTA, CTB, 64);
  fft2(CTB, CTB, 64, +1.0f, 1);
  k_real<<<EL(S[4]), 256, 0, stream>>>(CTB, RECON, S[4]);
  autocorr(RECON, 64, O_AC + NS, NS + 1);
  reduceArr(RECON, S[4], 0);
  k_fin_skewkurt<<<1, 1, 0, stream>>>(STATS, out, (float)S[4], NS);

  for (int s = NS - 1; s >= 0; --s) {
    int N = Nsz[s], n = S[s], Nh = N >> 1;
    for (int b = 0; b < K; ++b)
      autocorr(MAGd[s] + (size_t)b * n, N, O_ACM + s * K + b, NS * K);
    // expand recon (net scale 1: x4 pad factor cancels /4)
    k_pack<<<EL(Nh * Nh), 256, 0, stream>>>(RECON, CTA, Nh * Nh);
    fft2(CTA, CTA, Nh, -1.0f, 0);
    k_expand<<<EL(n), 256, 0, stream>>>(CTA, CTB, Nh);
    fft2(CTB, CTB, N, +1.0f, 1);
    k_real<<<EL(n), 256, 0, stream>>>(CTB, RECON, n);
    // recon_band accumulate
    k_zero<<<EL(2 * n), 256, 0, stream>>>((float*)RDFT, 2 * n);
    for (int b = 0; b < K; ++b) {
      k_pack<<<EL(n), 256, 0, stream>>>(REALCd[s] + (size_t)b * n, CTA, n);
      fft2(CTA, CTA, N, -1.0f, 0);
      k_recon_accum<<<EL(n), 256, 0, stream>>>(CTA, RDFT, N, b);
    }
    k_mask_lo0<<<EL(n), 256, 0, stream>>>(RDFT, RDFT, N);
    fft2(RDFT, RDFT, N, +1.0f, 1);
    k_real_add<<<EL(n), 256, 0, stream>>>(RDFT, RECON, n);
    autocorr(RECON, N, O_AC + s, NS + 1);
    reduceArr(RECON, n, 0);
    k_fin_skewkurt<<<1, 1, 0, stream>>>(STATS, out, (float)n, s);
  }

  // ---- cross-correlations (WMMA tall-skinny A^T B) ----
  for (int s = 0; s < NS; ++s) {
    int N = Nsz[s], n = S[s];
    int nrp;
    if (s < NS - 1) {
      nrp = 2 * K;  // 8
      int Nh = N >> 1;
      for (int b = 0; b < K; ++b) {
        fft2(BAND[s + 1][b], CTA, Nh, -1.0f, 0);
        k_expand<<<EL(n), 256, 0, stream>>>(CTA, CTB, Nh);
        fft2(CTB, CTB, N, +1.0f, 1);
        k_mag<<<EL(n), 256, 0, stream>>>(CTB, MGT, n);
        reduceArr(MGT, n, 0);
        k_fin_mean<<<1, 1, 0, stream>>>(STATS, out, (float)n, -1);
        k_parent_cols<<<EL(n), 256, 0, stream>>>(CTB, STATS,
                                                 PARM + (size_t)b * n,
                                                 RPAR + (size_t)b * n,
                                                 RPAR + (size_t)(K + b) * n, n);
      }
    } else {
      nrp = 5;
      k_pack<<<EL(S[4]), 256, 0, stream>>>(RLP, CTA, S[4]);
      fft2(CTA, CTA, 64, -1.0f, 0);
      k_expand<<<EL(n), 256, 0, stream>>>(CTA, CTB, 64);
      fft2(CTB, CTB, N, +1.0f, 1);
      k_real<<<EL(n), 256, 0, stream>>>(CTB, MGT, n);
      k_roll5<<<EL(n), 256, 0, stream>>>(MGT, RPAR, N);
    }
    float invn = 1.0f / (float)n;
    // coc_mag[:,:,s]
    k_zero<<<1, 256, 0, stream>>>(CACC, 256);
    k_wmma_corr<<<256, 32, 0, stream>>>(MAGd[s], MAGd[s], n, K, K, CACC);
    k_corr_write<<<1, 256, 0, stream>>>(CACC, out, K, K, K, NS + 1, O_COCM + s, invn);
    // csc_mag[:,:,s]
    if (s < NS - 1) {
      k_zero<<<1, 256, 0, stream>>>(CACC, 256);
      k_wmma_corr<<<256, 32, 0, stream>>>(MAGd[s], PARM, n, K, K, CACC);
      k_corr_write<<<1, 256, 0, stream>>>(CACC, out, K, K, K, NS, O_CSCM + s, invn);
    }
    // coc_real[:K,:K,s]
    k_zero<<<1, 256, 0, stream>>>(CACC, 256);
    k_wmma_corr<<<256, 32, 0, stream>>>(REALCd[s], REALCd[s], n, K, K, CACC);
    k_corr_write<<<1, 256, 0, stream>>>(CACC, out, K, K, 2 * K, NS + 1, O_COCR + s, invn);
    // csc_real[:K,:nrp,s]
    k_zero<<<1, 256, 0, stream>>>(CACC, 256);
    k_wmma_corr<<<256, 32, 0, stream>>>(REALCd[s], RPAR, n, K, nrp, CACC);
    k_corr_write<<<1, 256, 0, stream>>>(CACC, out, K, nrp, 2 * K, NS, O_CSCR + s, invn);
    // coc_real[:nrp,:nrp,NS] for final scale
    if (s == NS - 1) {
      k_zero<<<1, 256, 0, stream>>>(CACC, 256);
      k_wmma_corr<<<256, 32, 0, stream>>>(RPAR, RPAR, n, nrp, nrp, CACC);
      k_corr_write<<<1, 256, 0, stream>>>(CACC, out, nrp, nrp, 2 * K, NS + 1,
                                          O_COCR + NS, 4.0f / (float)n);
    }
  }

  // ---- var_hp ----
  reduceArr(RHP, S[0], 0);
  k_fin_varhp<<<1, 1, 0, stream>>>(STATS, out, (float)S[0]);
}